// DST_16509854286143
// MI455X (gfx1250) — compile-verified
//
#include <hip/hip_runtime.h>
#include <math.h>

// ---------------------------------------------------------------------------
// CDNA5 f32 WMMA: D(16x16 f32) = A(16x4 f32) x B(4x16 f32) + C
// A frag: lane L holds M = L%16, VGPR v holds K = k0 + v + 2*(L/16)
// B frag: lane L holds N = L%16, VGPR v holds K = k0 + v + 2*(L/16)
// C frag: lane L holds N = L%16, VGPR r holds M = r + 8*(L/16)
// ---------------------------------------------------------------------------
typedef __attribute__((ext_vector_type(2))) float v2f;
typedef __attribute__((ext_vector_type(8))) float v8f;

__device__ __forceinline__ v8f wmma_f32(v2f a, v2f b, v8f c) {
  return __builtin_amdgcn_wmma_f32_16x16x4_f32(false, a, false, b, (short)0, c,
                                               false, false);
}

// ===========================================================================
// Implicit-GEMM convolution.  out[b][oc][oh][ow] = sum_{ic,kh,kw} in * w
// GEMM view: M = oc, N = (b,oh,ow) pixel, K = ic*KH*KW (OIHW weights are
// exactly the row-major A matrix).  One wave computes a 16(M) x 16(N) tile.
// ===========================================================================
template <int CIN, int COUT, int KH, int KW, int STR, int PAD, bool PRE_RELU,
          bool RELU_OUT, bool HAS_BIAS, bool HAS_BN, bool ADD_RES>
__global__ __launch_bounds__(128) void dst_conv_wmma_k(
    const float* __restrict__ in, const float* __restrict__ wgt,
    const float* __restrict__ bias, const float* __restrict__ bns,
    const float* __restrict__ bnb, const float* __restrict__ resid,
    float* __restrict__ out, int B, int Hin, int Win, int Hout, int Wout) {
  constexpr int K = CIN * KH * KW;
  const int lane = threadIdx.x & 31;
  const int wv = threadIdx.x >> 5;
  const int hi = lane >> 4;
  const int l16 = lane & 15;
  const int tileM = blockIdx.y;
  const long long tileN = (long long)blockIdx.x * 4 + wv;
  const int HW = Hout * Wout;
  const long long Npix = (long long)B * HW;
  if (tileN * 16 >= Npix) return;

  // Per-lane output pixel (column of B / C)
  const long long n = tileN * 16 + l16;
  const int b = (int)(n / HW);
  const int hw = (int)(n % HW);
  const int oh = hw / Wout;
  const int ow = hw % Wout;
  const int ih0 = oh * STR - PAD;
  const int iw0 = ow * STR - PAD;
  const float* inB = in + (long long)b * CIN * Hin * Win;
  const int mrow = tileM * 16 + l16;  // A row for this lane

  v8f c = {0.f, 0.f, 0.f, 0.f, 0.f, 0.f, 0.f, 0.f};
  for (int k0 = 0; k0 < K; k0 += 4) {
    v2f a, bf;
#pragma unroll
    for (int v = 0; v < 2; ++v) {
      const int k = k0 + v + 2 * hi;
      a[v] = wgt[(long long)mrow * K + k];
      const int kw = k % KW;
      const int t = k / KW;
      const int kh = t % KH;
      const int ic = t / KH;
      const int ih = ih0 + kh;
      const int iw = iw0 + kw;
      float val = 0.f;
      if (ih >= 0 && ih < Hin && iw >= 0 && iw < Win)
        val = inB[((long long)ic * Hin + ih) * Win + iw];
      if (PRE_RELU) val = val > 0.f ? val : 0.f;
      bf[v] = val;
    }
    c = wmma_f32(a, bf, c);
  }

  float* outB = out + (long long)b * COUT * HW + hw;
  const float* resB = ADD_RES ? (resid + (long long)b * COUT * HW + hw) : out;
#pragma unroll
  for (int r = 0; r < 8; ++r) {
    const int oc = tileM * 16 + r + 8 * hi;
    float v = c[r];
    if (HAS_BIAS) v += bias[oc];
    if (HAS_BN) v = v * bns[oc] + bnb[oc];
    if (ADD_RES) v += resB[(long long)oc * HW];
    if (RELU_OUT) v = v > 0.f ? v : 0.f;
    outB[(long long)oc * HW] = v;
  }
}

// ===========================================================================
// Implicit-GEMM ConvTranspose2d (torch semantics), k=4, s=2, p=1.
// Weight layout (CIN, COUT, 4, 4).  Output tap valid when (o+1-k) is even.
// ===========================================================================
template <int CIN, int COUT, bool PRE_RELU, bool RELU_OUT>
__global__ __launch_bounds__(128) void dst_convt_wmma_k(
    const float* __restrict__ in, const float* __restrict__ wgt,
    const float* __restrict__ bias, float* __restrict__ out, int B, int Hin,
    int Win, int Hout, int Wout) {
  constexpr int K = CIN * 16;
  const int lane = threadIdx.x & 31;
  const int wv = threadIdx.x >> 5;
  const int hi = lane >> 4;
  const int l16 = lane & 15;
  const int tileM = blockIdx.y;
  const long long tileN = (long long)blockIdx.x * 4 + wv;
  const int HW = Hout * Wout;
  const long long Npix = (long long)B * HW;
  if (tileN * 16 >= Npix) return;

  const long long n = tileN * 16 + l16;
  const int b = (int)(n / HW);
  const int hw = (int)(n % HW);
  const int oh = hw / Wout;
  const int ow = hw % Wout;
  const float* inB = in + (long long)b * CIN * Hin * Win;
  const int oc_a = tileM * 16 + l16;  // A row

  v8f c = {0.f, 0.f, 0.f, 0.f, 0.f, 0.f, 0.f, 0.f};
  for (int k0 = 0; k0 < K; k0 += 4) {
    v2f a, bf;
#pragma unroll
    for (int v = 0; v < 2; ++v) {
      const int k = k0 + v + 2 * hi;
      const int kw = k & 3;
      const int kh = (k >> 2) & 3;
      const int ic = k >> 4;
      a[v] = wgt[((long long)ic * COUT + oc_a) * 16 + kh * 4 + kw];
      const int ihn = oh + 1 - kh;
      const int iwn = ow + 1 - kw;
      float val = 0.f;
      if (ihn >= 0 && iwn >= 0 && !(ihn & 1) && !(iwn & 1)) {
        const int ih = ihn >> 1, iw = iwn >> 1;
        if (ih < Hin && iw < Win)
          val = inB[((long long)ic * Hin + ih) * Win + iw];
      }
      if (PRE_RELU) val = val > 0.f ? val : 0.f;
      bf[v] = val;
    }
    c = wmma_f32(a, bf, c);
  }

  float* outB = out + (long long)b * COUT * HW + hw;
#pragma unroll
  for (int r = 0; r < 8; ++r) {
    const int oc = tileM * 16 + r + 8 * hi;
    float v = c[r] + bias[oc];
    if (RELU_OUT) v = v > 0.f ? v : 0.f;
    outB[(long long)oc * HW] = v;
  }
}

// ===========================================================================
// VQ: top-3 nearest codes per pixel, score = ||e||^2 - 2 f.e  (||f||^2 is
// constant per row -> irrelevant for argmin).  WMMA GEMM: A = 16 pixels x
// 128 dims (resident in 64 VGPRs), B = 128 x 16 codes per tile, 64 tiles.
// ===========================================================================
__device__ __forceinline__ void top3_ins(float d, int i, float* bd, int* bi) {
  if (d < bd[2]) {
    if (d < bd[1]) {
      bd[2] = bd[1]; bi[2] = bi[1];
      if (d < bd[0]) { bd[1] = bd[0]; bi[1] = bi[0]; bd[0] = d; bi[0] = i; }
      else           { bd[1] = d;     bi[1] = i; }
    } else { bd[2] = d; bi[2] = i; }
  }
}

__global__ __launch_bounds__(128) void dst_vq_top3_k(
    const float* __restrict__ z, const float* __restrict__ emb,
    const float* __restrict__ enorm, int* __restrict__ idx0,
    int* __restrict__ idx2) {
  const int lane = threadIdx.x & 31;
  const int wv = threadIdx.x >> 5;
  const int hi = lane >> 4;
  const int l16 = lane & 15;
  const int pixBase = (blockIdx.x * 4 + wv) * 16;

  // A fragment: pixel = pixBase + l16, dim striped per layout; keep resident.
  const int p = pixBase + l16;
  const int b = p >> 12;       // / (64*64)
  const int hw = p & 4095;
  const float* zB = z + (long long)b * 128 * 4096 + hw;
  v2f a[32];
#pragma unroll
  for (int k0 = 0; k0 < 32; ++k0) {
#pragma unroll
    for (int v = 0; v < 2; ++v) {
      const int d = k0 * 4 + v + 2 * hi;
      a[k0][v] = zB[(long long)d * 4096];
    }
  }

  float bd[8][3];
  int bi[8][3];
#pragma unroll
  for (int r = 0; r < 8; ++r) {
    bd[r][0] = bd[r][1] = bd[r][2] = 3.0e38f;
    bi[r][0] = bi[r][1] = bi[r][2] = 0;
  }

  for (int ct = 0; ct < 64; ++ct) {
    const int code = ct * 16 + l16;
    const float en = enorm[code];
    const float* eB = emb + (long long)code * 128;
    v8f c = {0.f, 0.f, 0.f, 0.f, 0.f, 0.f, 0.f, 0.f};
#pragma unroll
    for (int k0 = 0; k0 < 32; ++k0) {
      v2f bf;
#pragma unroll
      for (int v = 0; v < 2; ++v) bf[v] = eB[k0 * 4 + v + 2 * hi];
      c = wmma_f32(a[k0], bf, c);
    }
#pragma unroll
    for (int r = 0; r < 8; ++r)
      top3_ins(en - 2.f * c[r], code, bd[r], bi[r]);
  }

  // Butterfly-merge the 16 lanes sharing each pixel (wave32, width-16 xor).
#pragma unroll
  for (int off = 8; off >= 1; off >>= 1) {
#pragma unroll
    for (int r = 0; r < 8; ++r) {
#pragma unroll
      for (int j = 0; j < 3; ++j) {
        float od = __shfl_xor(bd[r][j], off, 16);
        int oi = __shfl_xor(bi[r][j], off, 16);
        top3_ins(od, oi, bd[r], bi[r]);
      }
    }
  }
  if (l16 == 0) {
#pragma unroll
    for (int r = 0; r < 8; ++r) {
      const int pix = pixBase + r + 8 * hi;
      idx0[pix] = bi[r][0];
      idx2[pix] = bi[r][2];
    }
  }
}

__global__ void dst_enorm_k(const float* __restrict__ emb,
                            float* __restrict__ enorm) {
  const int k = blockIdx.x * blockDim.x + threadIdx.x;
  if (k < 1024) {
    const float* e = emb + (long long)k * 128;
    float s = 0.f;
#pragma unroll 8
    for (int d = 0; d < 128; ++d) s += e[d] * e[d];
    enorm[k] = s;
  }
}

// quant (NCHW) = emb[idx0], plus partial sum of (q - z)^2 for the loss.
__global__ __launch_bounds__(256) void dst_vq_gather_k(
    const float* __restrict__ z, const float* __restrict__ emb,
    const int* __restrict__ idx0, float* __restrict__ quant,
    float* __restrict__ loss_acc) {
  __shared__ float red[256];
  const long long gid = (long long)blockIdx.x * 256 + threadIdx.x;  // 8.4M
  const int p = (int)(gid >> 7);
  const int d = (int)(gid & 127);
  const int b = p >> 12;
  const int hw = p & 4095;
  const long long zi = ((long long)b * 128 + d) * 4096 + hw;
  const float q = emb[(long long)idx0[p] * 128 + d];
  const float zz = z[zi];
  quant[zi] = q;
  const float df = q - zz;
  red[threadIdx.x] = df * df;
  __syncthreads();
  for (int s = 128; s > 0; s >>= 1) {
    if (threadIdx.x < s) red[threadIdx.x] += red[threadIdx.x + s];
    __syncthreads();
  }
  if (threadIdx.x == 0) atomicAdd(loss_acc, red[0]);
}

__global__ void dst_vq_hist_k(const int* __restrict__ idx2,
                              float* __restrict__ counts) {
  const int p = blockIdx.x * blockDim.x + threadIdx.x;
  if (p < 65536) atomicAdd(&counts[idx2[p]], 1.0f);
}

__global__ __launch_bounds__(256) void dst_vq_final_k(
    const float* __restrict__ counts, const float* __restrict__ loss_acc,
    float* __restrict__ out, int out_size) {
  __shared__ float red[256];
  float s = 0.f;
  for (int k = threadIdx.x; k < 1024; k += 256) {
    const float pr = counts[k] * (1.f / 65536.f);
    s += pr * logf(pr + 1e-10f);
  }
  red[threadIdx.x] = s;
  __syncthreads();
  for (int t = 128; t > 0; t >>= 1) {
    if (threadIdx.x < t) red[threadIdx.x] += red[threadIdx.x + t];
    __syncthreads();
  }
  if (threadIdx.x == 0) {
    out[0] = 0.25f * loss_acc[0] / (65536.f * 128.f);
    out[out_size - 1] = expf(-red[0]);
  }
}

__global__ void dst_init_k(float* counts, float* loss_acc) {
  const int t = blockIdx.x * blockDim.x + threadIdx.x;
  if (t < 1024) counts[t] = 0.f;
  if (t == 0) loss_acc[0] = 0.f;
}

// Eval-mode BN -> per-channel scale/shift
__global__ void dst_bn_prep_k(const float* g, const float* b, const float* m,
                              const float* v, int n, float* scale,
                              float* shift) {
  const int t = blockIdx.x * blockDim.x + threadIdx.x;
  if (t < n) {
    const float inv = g[t] * rsqrtf(v[t] + 1e-5f);
    scale[t] = inv;
    shift[t] = b[t] - m[t] * inv;
  }
}

// Final ConvTranspose2d 64->1 (k4 s2 p1), direct (Cout==1 => no GEMM win).
__global__ __launch_bounds__(256) void dst_convt2_k(
    const float* __restrict__ in, const float* __restrict__ wgt,
    const float* __restrict__ bias, float* __restrict__ out) {
  const long long gid = (long long)blockIdx.x * 256 + threadIdx.x;  // 1048576
  const int ow = (int)(gid & 255);
  const int oh = (int)((gid >> 8) & 255);
  const int b = (int)(gid >> 16);
  float acc = bias[0];
#pragma unroll
  for (int kh = 0; kh < 4; ++kh) {
    const int ihn = oh + 1 - kh;
    if (ihn < 0 || (ihn & 1)) continue;
    const int ih = ihn >> 1;
    if (ih >= 128) continue;
#pragma unroll
    for (int kw = 0; kw < 4; ++kw) {
      const int iwn = ow + 1 - kw;
      if (iwn < 0 || (iwn & 1)) continue;
      const int iw = iwn >> 1;
      if (iw >= 128) continue;
      const float* ib = in + (long long)b * 64 * 16384 + ih * 128 + iw;
      const float* wp = wgt + kh * 4 + kw;
      float s = 0.f;
#pragma unroll 4
      for (int ic = 0; ic < 64; ++ic) s += ib[(long long)ic * 16384] * wp[ic * 16];
      acc += s;
    }
  }
  out[gid] = acc;
}

// ===========================================================================
// Host side
// ===========================================================================
extern "C" void kernel_launch(void* const* d_in, const int* in_sizes, int n_in,
                              void* d_out, int out_size, void* d_ws,
                              size_t ws_size, hipStream_t stream) {
  (void)in_sizes; (void)n_in; (void)ws_size;
  // ---- input indices (depth-first, dict insertion order of setup_inputs) --
  enum {
    IX = 0,
    P_C1W, P_C1B, P_C2W, P_C2B, P_C3W, P_C3B,
    ER0,                 // proj res[0]: w1,bn1_g,bn1_b,bn1_m,bn1_v,w2,bn2_g,bn2_b,bn2_m,bn2_v
    ER1 = ER0 + 10,
    PREVQ_W = ER1 + 10, PREVQ_B, EMB,
    D_C1W, D_C1B,
    DR0,
    DR1 = DR0 + 10,
    T1W = DR1 + 10, T1B, T2W, T2B
  };
  const float* F(const void*);
  auto f = [&](int i) { return (const float*)d_in[i]; };

  // ---- workspace carve (floats) ------------------------------------------
  float* ws = (float*)d_ws;
  size_t o = 0;
  float* A0 = ws + o; o += 16777216;  // 16x64x128x128 (conv1 out / t1 out)
  float* A2 = ws + o; o += 8388608;   // 16x128x64x64 ping
  float* A3 = ws + o; o += 2097152;   // 16x32x64x64 res hidden
  float* A4 = ws + o; o += 8388608;   // 16x128x64x64 pong
  float* A5 = ws + o; o += 8388608;   // z
  float* A6 = ws + o; o += 8388608;   // quant
  int*   idx0 = (int*)(ws + o); o += 65536;
  int*   idx2 = (int*)(ws + o); o += 65536;
  float* enorm = ws + o; o += 1024;
  float* counts = ws + o; o += 1024;
  float* loss_acc = ws + o; o += 16;
  float* bnw = ws + o; o += 4 * 320;  // per block: s1(32) h1(32) s2(128) h2(128)

  float* out = (float*)d_out;

  dst_init_k<<<4, 256, 0, stream>>>(counts, loss_acc);

  // ---- BN precompute for the 4 residual blocks ---------------------------
  const int resIdx[4] = {ER0, ER1, DR0, DR1};
  for (int j = 0; j < 4; ++j) {
    const int r = resIdx[j];
    float* bb = bnw + j * 320;
    dst_bn_prep_k<<<1, 128, 0, stream>>>(f(r + 1), f(r + 2), f(r + 3), f(r + 4),
                                         32, bb + 0, bb + 32);
    dst_bn_prep_k<<<1, 128, 0, stream>>>(f(r + 6), f(r + 7), f(r + 8), f(r + 9),
                                         128, bb + 64, bb + 192);
  }

  // ---- Encoder -----------------------------------------------------------
  // conv1: 1->64, 4x4 s2 p1, +bias, ReLU.  256->128.  Npix=262144
  dst_conv_wmma_k<1, 64, 4, 4, 2, 1, false, true, true, false, false>
      <<<dim3(4096, 4), 128, 0, stream>>>(f(IX), f(P_C1W), f(P_C1B), nullptr,
                                          nullptr, nullptr, A0, 16, 256, 256,
                                          128, 128);
  // conv2: 64->128, 4x4 s2 p1, +bias, ReLU.  128->64.  Npix=65536
  dst_conv_wmma_k<64, 128, 4, 4, 2, 1, false, true, true, false, false>
      <<<dim3(1024, 8), 128, 0, stream>>>(A0, f(P_C2W), f(P_C2B), nullptr,
                                          nullptr, nullptr, A2, 16, 128, 128,
                                          64, 64);
  // conv3: 128->128, 3x3 p1, +bias.
  dst_conv_wmma_k<128, 128, 3, 3, 1, 1, false, false, true, false, false>
      <<<dim3(1024, 8), 128, 0, stream>>>(A2, f(P_C3W), f(P_C3B), nullptr,
                                          nullptr, nullptr, A4, 16, 64, 64, 64,
                                          64);

  auto res_block = [&](const float* xin, float* xout, int widx, float* bb) {
    // conv3x3 128->32 on relu(x), +BN1, ReLU out
    dst_conv_wmma_k<128, 32, 3, 3, 1, 1, true, true, false, true, false>
        <<<dim3(1024, 2), 128, 0, stream>>>(xin, f(widx), nullptr, bb + 0,
                                            bb + 32, nullptr, A3, 16, 64, 64,
                                            64, 64);
    // conv1x1 32->128, +BN2, + residual x
    dst_conv_wmma_k<32, 128, 1, 1, 1, 0, false, false, false, true, true>
        <<<dim3(1024, 8), 128, 0, stream>>>(A3, f(widx + 5), nullptr, bb + 64,
                                            bb + 192, xin, xout, 16, 64, 64,
                                            64, 64);
  };

  res_block(A4, A2, ER0, bnw + 0 * 320);
  res_block(A2, A4, ER1, bnw + 1 * 320);

  // pre-VQ 1x1: relu fused (res_stack final ReLU), 128->128, +bias -> z
  dst_conv_wmma_k<128, 128, 1, 1, 1, 0, true, false, true, false, false>
      <<<dim3(1024, 8), 128, 0, stream>>>(A4, f(PREVQ_W), f(PREVQ_B), nullptr,
                                          nullptr, nullptr, A5, 16, 64, 64, 64,
                                          64);

  // ---- VQ ----------------------------------------------------------------
  dst_enorm_k<<<4, 256, 0, stream>>>(f(EMB), enorm);
  dst_vq_top3_k<<<1024, 128, 0, stream>>>(A5, f(EMB), enorm, idx0, idx2);
  dst_vq_gather_k<<<32768, 256, 0, stream>>>(A5, f(EMB), idx0, A6, loss_acc);
  dst_vq_hist_k<<<256, 256, 0, stream>>>(idx2, counts);

  // ---- Decoder -----------------------------------------------------------
  dst_conv_wmma_k<128, 128, 3, 3, 1, 1, false, false, true, false, false>
      <<<dim3(1024, 8), 128, 0, stream>>>(A6, f(D_C1W), f(D_C1B), nullptr,
                                          nullptr, nullptr, A2, 16, 64, 64, 64,
                                          64);
  res_block(A2, A4, DR0, bnw + 2 * 320);
  res_block(A4, A2, DR1, bnw + 3 * 320);

  // t1: ConvT 128->64, relu fused on input (res_stack ReLU) and output.
  dst_convt_wmma_k<128, 64, true, true>
      <<<dim3(4096, 4), 128, 0, stream>>>(A2, f(T1W), f(T1B), A0, 16, 64, 64,
                                          128, 128);
  // t2: ConvT 64->1 -> x_recon at out[1..1048576]
  dst_convt2_k<<<4096, 256, 0, stream>>>(A0, f(T2W), f(T2B), out + 1);

  // loss (out[0]) + perplexity (out[out_size-1])
  dst_vq_final_k<<<1, 256, 0, stream>>>(counts, loss_acc, out, out_size);
}